// TVRestorer_31971736552101
// MI455X (gfx1250) — compile-verified
//
#include <hip/hip_runtime.h>
#include <cmath>

// ---------------------------------------------------------------------------
// TV restorer (FFT-split-Bregman style) for (4,3,256,256) f32.
// FFT2 implemented as DFT-matrix GEMMs on the CDNA5 WMMA f32 pipe:
//   fft2(X)  = W  @ X  @ W          (W[u,y] = e^{-2pi i u y/256}, symmetric)
//   ifft2(F) = WI @ F  @ WI         (WI = conj(W)/256  -> total 1/N^2)
// Everything else is elementwise (FDH/FDV/FD2 are analytic).
// ---------------------------------------------------------------------------

typedef float v2f __attribute__((ext_vector_type(2)));
typedef float v8f __attribute__((ext_vector_type(8)));

#define NPIX   65536            // 256*256 (one plane, also DFT matrix size)
#define PLANES 12               // 4*3
#define STOT   (PLANES * NPIX)  // elements per full real field
#define ALPHA  1e-6f

// ---------------------------------------------------------------------------
// Batched complex GEMM  C[p] = A[p] * B[p], all matrices 256x256 row-major.
// stride==0 -> operand shared across planes. One wave32 computes one 32x16
// output tile (two 16x16 M-blocks sharing the B fragment) via
// V_WMMA_F32_16X16X4_F32 (K-loop step 4, 64 steps).
// Complex product via real accumulators; combine at store:
//   Cre = accRR - accII ; Cim = accRI + accIR
// Template flags drop unused imaginary work (real input / real-only output).
// ---------------------------------------------------------------------------
template <bool AIM, bool BIM, bool CIM>
__global__ __launch_bounds__(128, 2) void cgemm_wmma(
    const float* __restrict__ Are, const float* __restrict__ Aim, int sA,
    const float* __restrict__ Bre, const float* __restrict__ Bim, int sB,
    float* __restrict__ Cre, float* __restrict__ Cim, int sC)
{
    const int lane   = threadIdx.x & 31;
    const int wave   = threadIdx.x >> 5;
    const int tile   = (blockIdx.x << 2) | wave;   // 0..127
    const int tm     = tile >> 4;                  // 0..7  (32-row block)
    const int tn     = tile & 15;                  // 0..15 (16-col block)
    const int plane  = blockIdx.y;
    const int laneLo = lane & 15;
    const int laneHi = lane >> 4;

    const float* are = Are + (size_t)plane * (size_t)sA;
    const float* aim = AIM ? (Aim + (size_t)plane * (size_t)sA) : nullptr;
    const float* bre = Bre + (size_t)plane * (size_t)sB;
    const float* bim = BIM ? (Bim + (size_t)plane * (size_t)sB) : nullptr;

    const int rowA0 = tm * 32 + laneLo;  // first M-block row this lane supplies
    const int rowA1 = rowA0 + 16;        // second M-block
    const int colB  = tn * 16 + laneLo;  // B col supplied by this lane

    v8f accRR0 = {}, accRR1 = {};
    v8f accII0 = {}, accII1 = {};
    v8f accRI0 = {}, accRI1 = {};
    v8f accIR0 = {}, accIR1 = {};

    for (int k = 0; k < 256; k += 4) {
        const int ka = k + laneHi * 2;   // lanes 0-15: K={k,k+1}; 16-31: {k+2,k+3}

        // A fragments: 16x4 each, row per lane, 2 consecutive K (b64 loads)
        v2f ar0 = *(const v2f*)(are + rowA0 * 256 + ka);
        v2f ar1 = *(const v2f*)(are + rowA1 * 256 + ka);
        // B fragment: 4x16, col per lane, 2 K values (stride-256 b32 loads)
        v2f br;
        br.x = bre[(ka    ) * 256 + colB];
        br.y = bre[(ka + 1) * 256 + colB];

        accRR0 = __builtin_amdgcn_wmma_f32_16x16x4_f32(
            false, ar0, false, br, (short)0, accRR0, false, false);
        accRR1 = __builtin_amdgcn_wmma_f32_16x16x4_f32(
            false, ar1, false, br, (short)0, accRR1, false, false);

        if constexpr (AIM) {
            v2f ai0 = *(const v2f*)(aim + rowA0 * 256 + ka);
            v2f ai1 = *(const v2f*)(aim + rowA1 * 256 + ka);
            if constexpr (CIM) {
                accIR0 = __builtin_amdgcn_wmma_f32_16x16x4_f32(
                    false, ai0, false, br, (short)0, accIR0, false, false);
                accIR1 = __builtin_amdgcn_wmma_f32_16x16x4_f32(
                    false, ai1, false, br, (short)0, accIR1, false, false);
            }
            if constexpr (BIM) {
                v2f bi;
                bi.x = bim[(ka    ) * 256 + colB];
                bi.y = bim[(ka + 1) * 256 + colB];
                accII0 = __builtin_amdgcn_wmma_f32_16x16x4_f32(
                    false, ai0, false, bi, (short)0, accII0, false, false);
                accII1 = __builtin_amdgcn_wmma_f32_16x16x4_f32(
                    false, ai1, false, bi, (short)0, accII1, false, false);
                if constexpr (CIM) {
                    accRI0 = __builtin_amdgcn_wmma_f32_16x16x4_f32(
                        false, ar0, false, bi, (short)0, accRI0, false, false);
                    accRI1 = __builtin_amdgcn_wmma_f32_16x16x4_f32(
                        false, ar1, false, bi, (short)0, accRI1, false, false);
                }
            }
        }
    }

    float* cre = Cre + (size_t)plane * (size_t)sC;
    float* cim = CIM ? (Cim + (size_t)plane * (size_t)sC) : nullptr;
    const int col   = tn * 16 + laneLo;
    const int row0a = tm * 32 + laneHi * 8;        // M-block 0
    const int row0b = row0a + 16;                  // M-block 1
#pragma unroll
    for (int r = 0; r < 8; ++r) {
        float vr0 = accRR0[r];
        float vr1 = accRR1[r];
        if constexpr (AIM && BIM) { vr0 -= accII0[r]; vr1 -= accII1[r]; }
        cre[(row0a + r) * 256 + col] = vr0;
        cre[(row0b + r) * 256 + col] = vr1;
        if constexpr (CIM) {
            float vi0 = 0.f, vi1 = 0.f;
            if constexpr (BIM) { vi0 += accRI0[r]; vi1 += accRI1[r]; }
            if constexpr (AIM) { vi0 += accIR0[r]; vi1 += accIR1[r]; }
            cim[(row0a + r) * 256 + col] = vi0;
            cim[(row0b + r) * 256 + col] = vi1;
        }
    }
}

// ---------------------------------------------------------------------------
// Elementwise kernels
// ---------------------------------------------------------------------------
__device__ inline float fd2_at(int y, int x) {
    // |1-e^{-2pi i y/256}|^2 + |1-e^{-2pi i x/256}|^2 + 1e-8
    float cy = cospif((float)y * (1.0f / 128.0f));
    float cx = cospif((float)x * (1.0f / 128.0f));
    return 4.0f - 2.0f * cy - 2.0f * cx + 1e-8f;
}

__global__ void k_init_w(float* __restrict__ wre, float* __restrict__ wim,
                         float* __restrict__ wire, float* __restrict__ wiim)
{
    int i = blockIdx.x * blockDim.x + threadIdx.x;
    if (i >= NPIX) return;
    int u = i >> 8, y = i & 255;
    int t = (u * y) & 255;                       // exact periodic reduction
    float a = (float)t * (1.0f / 128.0f);        // angle / pi
    float c = cospif(a), s = sinpif(a);
    wre[i]  = c;                                 // e^{-i theta}
    wim[i]  = -s;
    wire[i] = c * (1.0f / 256.0f);               // conj(W)/256
    wiim[i] = s * (1.0f / 256.0f);
}

__global__ void k_zero(float* __restrict__ p, int n)
{
    int i = blockIdx.x * blockDim.x + threadIdx.x;
    if (i < n) p[i] = 0.0f;
}

// Loop-invariants: FkFSTy = conj(Fk)*FSTy ; den = down(Fk2/FD2)+alpha ;
// Fr0 = FkFSTy / (down(Fk2)+alpha).  down == mean of 4 half-shifted quadrants.
__global__ void k_precompute(const float* __restrict__ fkre, const float* __restrict__ fkim,
                             const float* __restrict__ fsre, const float* __restrict__ fsim,
                             float* __restrict__ kfre, float* __restrict__ kfim,
                             float* __restrict__ den,
                             float* __restrict__ f0re, float* __restrict__ f0im)
{
    int i = blockIdx.x * blockDim.x + threadIdx.x;
    if (i >= STOT) return;
    int r = i & (NPIX - 1);
    int base = i - r;
    int y = r >> 8, x = r & 255;

    float a = fkre[i], b = fkim[i];
    float fr = fsre[i], fi = fsim[i];
    float kr = a * fr + b * fi;                  // conj(Fk)*FSTy
    float ki = a * fi - b * fr;

    int y0 = y & 127, x0 = x & 127;
    float sf2 = 0.0f, srat = 0.0f;
#pragma unroll
    for (int dy = 0; dy < 2; ++dy) {
#pragma unroll
        for (int dx = 0; dx < 2; ++dx) {
            int yy = y0 + dy * 128, xx = x0 + dx * 128;
            int q = base + (yy << 8) + xx;
            float qa = fkre[q], qb = fkim[q];
            float f2 = qa * qa + qb * qb;
            sf2  += f2;
            srat += f2 / fd2_at(yy, xx);
        }
    }
    float down = 0.25f * sf2;
    den[i]  = 0.25f * srat + ALPHA;
    kfre[i] = kr;
    kfim[i] = ki;
    float inv = 1.0f / (down + ALPHA);
    f0re[i] = kr * inv;
    f0im[i] = ki * inv;
}

// Fr = FkFSTy + alpha*( conj(FDH)*G1 + conj(FDV)*G2 )
__global__ void k_fr1(const float* __restrict__ kfre, const float* __restrict__ kfim,
                      const float* __restrict__ g1re, const float* __restrict__ g1im,
                      const float* __restrict__ g2re, const float* __restrict__ g2im,
                      float* __restrict__ frre, float* __restrict__ frim)
{
    int i = blockIdx.x * blockDim.x + threadIdx.x;
    if (i >= STOT) return;
    int r = i & (NPIX - 1);
    int y = r >> 8, x = r & 255;
    float cx = cospif((float)x * (1.0f / 128.0f)), sx = sinpif((float)x * (1.0f / 128.0f));
    float cy = cospif((float)y * (1.0f / 128.0f)), sy = sinpif((float)y * (1.0f / 128.0f));
    float hr = 1.0f - cx, hi = -sx;              // conj(FDH)
    float vr = 1.0f - cy, vi = -sy;              // conj(FDV)
    float g1r = g1re[i], g1i = g1im[i];
    float g2r = g2re[i], g2i = g2im[i];
    float t_r = hr * g1r - hi * g1i + vr * g2r - vi * g2i;
    float t_i = hr * g1i + hi * g1r + vr * g2i + vi * g2r;
    frre[i] = kfre[i] + ALPHA * t_r;
    frim[i] = kfim[i] + ALPHA * t_i;
}

// Fr' = (Fr - conj(Fk)*down(Fk*Fr/FD2)/den) / (alpha*FD2);  also Fr'*FDH, Fr'*FDV
__global__ void k_fr2(const float* __restrict__ fkre, const float* __restrict__ fkim,
                      const float* __restrict__ frre, const float* __restrict__ frim,
                      const float* __restrict__ den,
                      float* __restrict__ fbre, float* __restrict__ fbim,
                      float* __restrict__ fhre, float* __restrict__ fhim,
                      float* __restrict__ fvre, float* __restrict__ fvim)
{
    int i = blockIdx.x * blockDim.x + threadIdx.x;
    if (i >= STOT) return;
    int r = i & (NPIX - 1);
    int base = i - r;
    int y = r >> 8, x = r & 255;
    int y0 = y & 127, x0 = x & 127;

    float sr = 0.0f, si = 0.0f;
#pragma unroll
    for (int dy = 0; dy < 2; ++dy) {
#pragma unroll
        for (int dx = 0; dx < 2; ++dx) {
            int yy = y0 + dy * 128, xx = x0 + dx * 128;
            int q = base + (yy << 8) + xx;
            float a = fkre[q], b = fkim[q];
            float u = frre[q], v = frim[q];
            float inv = 1.0f / fd2_at(yy, xx);
            sr += (a * u - b * v) * inv;         // Fk*Fr/FD2
            si += (a * v + b * u) * inv;
        }
    }
    sr *= 0.25f; si *= 0.25f;

    float a = fkre[i], b = fkim[i];
    float nr = a * sr + b * si;                  // conj(Fk)*g
    float ni = a * si - b * sr;
    float dv = den[i];
    float invf = 1.0f / (ALPHA * fd2_at(y, x));
    float rr = (frre[i] - nr / dv) * invf;
    float ri = (frim[i] - ni / dv) * invf;
    fbre[i] = rr; fbim[i] = ri;

    float cx = cospif((float)x * (1.0f / 128.0f)), sx = sinpif((float)x * (1.0f / 128.0f));
    float cy = cospif((float)y * (1.0f / 128.0f)), sy = sinpif((float)y * (1.0f / 128.0f));
    float hr = 1.0f - cx, hi = sx;               // FDH
    float vr = 1.0f - cy, vi = sy;               // FDV
    fhre[i] = rr * hr - ri * hi; fhim[i] = rr * hi + ri * hr;
    fvre[i] = rr * vr - ri * vi; fvim[i] = rr * vi + ri * vr;
}

// FISTA momentum + isotropic soft-threshold + Bregman update.
// Emits V1=U1-D1, V2=U2-D2 for the next iteration; writes pred to d_out.
__global__ void k_update(const float* __restrict__ xn, const float* __restrict__ xo,
                         const float* __restrict__ dh, const float* __restrict__ dvv,
                         float* __restrict__ d1, float* __restrict__ d2,
                         float* __restrict__ v1, float* __restrict__ v2,
                         float* __restrict__ outp, float c1, float gam)
{
    int i = blockIdx.x * blockDim.x + threadIdx.x;
    if (i >= STOT) return;
    float xv = xn[i], xold = xo[i];
    outp[i] = xv + c1 * (xv - xold);
    float hh = dh[i], vv = dvv[i];
    float a1 = hh + d1[i];
    float a2 = vv + d2[i];
    float nu = sqrtf(a1 * a1 + a2 * a2 + 1e-8f);
    float A = fmaxf(0.0f, nu - gam) + 1e-8f;
    A = A / (A + gam);
    float u1 = A * a1, u2 = A * a2;
    float nd1 = d1[i] + hh - u1;
    float nd2 = d2[i] + vv - u2;
    d1[i] = nd1; d2[i] = nd2;
    v1[i] = u1 - nd1; v2[i] = u2 - nd2;
}

// ---------------------------------------------------------------------------
// Host orchestration (graph-capturable: launches only, fixed sequence)
// ---------------------------------------------------------------------------
extern "C" void kernel_launch(void* const* d_in, const int* in_sizes, int n_in,
                              void* d_out, int out_size, void* d_ws, size_t ws_size,
                              hipStream_t stream)
{
    const float* STy  = (const float*)d_in[0];
    const float* Fkre = (const float*)d_in[1];
    const float* Fkim = (const float*)d_in[2];
    // d_in[3] is sf; reference setup uses sf==2 (downsample code assumes it).
    float* out = (float*)d_out;

    const size_t M = NPIX, S = STOT;
    if (ws_size < (4 * M + 25 * S) * sizeof(float)) return;

    float* ws   = (float*)d_ws;
    float* Wre  = ws;
    float* Wim  = ws + M;
    float* WIre = ws + 2 * M;
    float* WIim = ws + 3 * M;
    float* F    = ws + 4 * M;
    auto fld = [&](int i) { return F + (size_t)i * S; };
    float *Tre  = fld(0),  *Tim  = fld(1);
    float *G1re = fld(2),  *G1im = fld(3);
    float *G2re = fld(4),  *G2im = fld(5);
    float *KFre = fld(6),  *KFim = fld(7);
    float *Den  = fld(8);
    float *FAre = fld(9),  *FAim = fld(10);
    float *FBre = fld(11), *FBim = fld(12);
    float *FHre = fld(13), *FHim = fld(14);
    float *FVre = fld(15), *FVim = fld(16);
    float *X0   = fld(17), *X1   = fld(18);
    float *DH   = fld(19), *DV   = fld(20);
    float *V1   = fld(21), *V2   = fld(22);
    float *D1   = fld(23), *D2   = fld(24);

    dim3 gb(32, PLANES), bb(128);       // 32 blocks * 4 waves = 128 32x16 tiles/plane
    dim3 ge((STOT + 255) / 256), be(256);

    // fft2 of a real field:  T = W*X ; F = T*W
    auto fwd_real = [&](const float* xr, float* fre, float* fim) {
        cgemm_wmma<true, false, true><<<gb, bb, 0, stream>>>(
            Wre, Wim, 0, xr, nullptr, NPIX, Tre, Tim, NPIX);
        cgemm_wmma<true, true, true><<<gb, bb, 0, stream>>>(
            Tre, Tim, NPIX, Wre, Wim, 0, fre, fim, NPIX);
    };
    // ifft2 real part:  T = WI*F ; X = Re(T*WI)
    auto inv_real = [&](const float* fre, const float* fim, float* xr) {
        cgemm_wmma<true, true, true><<<gb, bb, 0, stream>>>(
            WIre, WIim, 0, fre, fim, NPIX, Tre, Tim, NPIX);
        cgemm_wmma<true, true, false><<<gb, bb, 0, stream>>>(
            Tre, Tim, NPIX, WIre, WIim, 0, xr, nullptr, NPIX);
    };

    // ---- prelude ----
    k_init_w<<<256, 256, 0, stream>>>(Wre, Wim, WIre, WIim);
    fwd_real(STy, G1re, G1im);                               // FSTy -> G1
    k_precompute<<<ge, be, 0, stream>>>(Fkre, Fkim, G1re, G1im,
                                        KFre, KFim, Den, FAre, FAim);
    inv_real(FAre, FAim, X0);                                // pred0
    k_zero<<<ge, be, 0, stream>>>(D1, STOT);
    k_zero<<<ge, be, 0, stream>>>(D2, STOT);

    float* xPrev = X0;
    float* xCur  = X1;
    const float* pv1 = X0;   // V1 = U1 - D1 = pred0 at k=0
    const float* pv2 = X0;

    double t = 1.0;
    for (int k = 0; k < 50; ++k) {
        fwd_real(pv1, G1re, G1im);
        fwd_real(pv2, G2re, G2im);
        k_fr1<<<ge, be, 0, stream>>>(KFre, KFim, G1re, G1im, G2re, G2im, FAre, FAim);
        k_fr2<<<ge, be, 0, stream>>>(Fkre, Fkim, FAre, FAim, Den,
                                     FBre, FBim, FHre, FHim, FVre, FVim);
        inv_real(FBre, FBim, xCur);                          // x_new
        inv_real(FHre, FHim, DH);                            // DHr
        inv_real(FVre, FVim, DV);                            // DVr

        double tn  = (1.0 + std::sqrt(1.0 + 4.0 * t * t)) * 0.5;
        float  c1  = (float)((t - 1.0) / tn);
        float  gam = (float)(0.1 * std::exp2(-(double)k / 50.0));
        t = tn;

        k_update<<<ge, be, 0, stream>>>(xCur, xPrev, DH, DV, D1, D2, V1, V2,
                                        out, c1, gam);

        float* tmp = xPrev; xPrev = xCur; xCur = tmp;
        pv1 = V1; pv2 = V2;
    }
}